// GCNet_20847771254912
// MI455X (gfx1250) — compile-verified
//
#include <hip/hip_runtime.h>

#define N_NODES 100000
#define N_EDGES 1600000

typedef __attribute__((ext_vector_type(2))) float v2f;
typedef __attribute__((ext_vector_type(8))) float v8f;

// ---------------------------------------------------------------------------
// Degree / normalization:  deg[c] = 1 (self loop) + #incoming edges at c,
// then dinv = rsqrt(deg).  norm(e) = dinv[row]*dinv[col]; self-loop = dinv^2.
// ---------------------------------------------------------------------------
__global__ void deg_init_kernel(float* __restrict__ deg) {
  int i = blockIdx.x * blockDim.x + threadIdx.x;
  if (i < N_NODES) deg[i] = 1.0f;  // self-loop contribution
}

__global__ void deg_count_kernel(float* __restrict__ deg,
                                 const int* __restrict__ col) {
  int e = blockIdx.x * blockDim.x + threadIdx.x;
  if (e < N_EDGES) {
    __hip_atomic_fetch_add(&deg[col[e]], 1.0f,
                           __ATOMIC_RELAXED, __HIP_MEMORY_SCOPE_AGENT);
  }
}

__global__ void deg_rsqrt_kernel(float* __restrict__ deg) {
  int i = blockIdx.x * blockDim.x + threadIdx.x;
  if (i < N_NODES) deg[i] = rsqrtf(deg[i]);  // deg >= 1 always
}

// ---------------------------------------------------------------------------
// Dense GEMM  Out[N, DOUT] = H[N, DIN] @ W[DIN, DOUT]   via V_WMMA_F32_16X16X4_F32.
// One wave computes a 16-row tile; K marched in steps of 4; DOUT covered by
// ceil(DOUT/16) N-tiles.  N_NODES = 6250 * 16 exactly, so no M remainder.
//
// f32 WMMA operand layout (wave32):
//   A 16x4 : lanes 0-15 -> M=lane, VGPR{0,1}=K{0,1}; lanes 16-31 -> K{2,3}
//   B 4x16 : lanes 0-15 -> N=lane, VGPR{0,1}=K{0,1}; lanes 16-31 -> K{2,3}
//   C/D    : VGPR v -> M = v (lanes 0-15) / v+8 (lanes 16-31), N = lane%16
// ---------------------------------------------------------------------------
template <int DIN, int DOUT>
__global__ void gemm_wmma_kernel(const float* __restrict__ H,
                                 const float* __restrict__ W,
                                 float* __restrict__ Out) {
  constexpr int NT = (DOUT + 15) / 16;  // 1 (DOUT=3) or 2 (DOUT=32)
  const int wave = (blockIdx.x * blockDim.x + threadIdx.x) >> 5;
  const int lane = threadIdx.x & 31;
  const int m0 = wave * 16;
  if (m0 >= N_NODES) return;  // uniform per wave: whole wave exits, EXEC stays full

  const int lhalf = lane & 15;
  const int khi = (lane >> 4) << 1;                    // 0 or 2
  const int ncl = (lhalf < DOUT) ? lhalf : (DOUT - 1); // clamp B column in-bounds

  const float* __restrict__ hrow = H + (size_t)(m0 + lhalf) * DIN;

  v8f acc0 = {};
  v8f acc1 = {};
#pragma unroll
  for (int k = 0; k < DIN; k += 4) {
    v2f a, b0;
    a.x = hrow[k + khi];
    a.y = hrow[k + khi + 1];
    b0.x = W[(k + khi) * DOUT + ncl];
    b0.y = W[(k + khi + 1) * DOUT + ncl];
    acc0 = __builtin_amdgcn_wmma_f32_16x16x4_f32(
        false, a, false, b0, (short)0, acc0, false, false);
    if (NT > 1) {
      v2f b1;
      b1.x = W[(k + khi) * DOUT + 16 + lhalf];
      b1.y = W[(k + khi + 1) * DOUT + 16 + lhalf];
      acc1 = __builtin_amdgcn_wmma_f32_16x16x4_f32(
          false, a, false, b1, (short)0, acc1, false, false);
    }
  }

  const int mrow = m0 + ((lane >> 4) << 3);
#pragma unroll
  for (int v = 0; v < 8; ++v) {
    const int m = mrow + v;
    if (lhalf < DOUT) Out[(size_t)m * DOUT + lhalf] = acc0[v];
    if (NT > 1)       Out[(size_t)m * DOUT + 16 + lhalf] = acc1[v];
  }
}

// ---------------------------------------------------------------------------
// Aggregation.  out[i,:] = hW[i,:] * dinv[i]^2 + bias   (self-loop + bias init)
// then per edge:  out[col,:] += hW[row,:] * dinv[row]*dinv[col]  (fp32 atomics,
// 32 consecutive lanes = 32 features of one edge -> coalesced rows, L2-resident)
// ---------------------------------------------------------------------------
template <int DOUT>
__global__ void agg_init_kernel(const float* __restrict__ hW,
                                const float* __restrict__ bias,
                                const float* __restrict__ dinv,
                                float* __restrict__ out) {
  long long t = (long long)blockIdx.x * blockDim.x + threadIdx.x;
  if (t >= (long long)N_NODES * DOUT) return;
  int i = (int)(t / DOUT);
  int f = (int)(t - (long long)i * DOUT);
  float d = dinv[i];
  out[t] = hW[t] * (d * d) + bias[f];
}

template <int DOUT>
__global__ void agg_edge_kernel(const float* __restrict__ hW,
                                const int* __restrict__ row,
                                const int* __restrict__ col,
                                const float* __restrict__ dinv,
                                float* __restrict__ out) {
  long long t = (long long)blockIdx.x * blockDim.x + threadIdx.x;
  if (t >= (long long)N_EDGES * DOUT) return;
  int e = (int)(t / DOUT);
  int f = (int)(t - (long long)e * DOUT);
  int r = row[e];
  int c = col[e];
  float w = dinv[r] * dinv[c];
  float v = hW[(size_t)r * DOUT + f] * w;
  __hip_atomic_fetch_add(&out[(size_t)c * DOUT + f], v,
                         __ATOMIC_RELAXED, __HIP_MEMORY_SCOPE_AGENT);
}

// ---------------------------------------------------------------------------
// Host launch.  Inputs (setup_inputs order):
//   [0] x [N,8]      [1] edge_index [2,E] (int32)
//   [2] W1 [8,32]  [3] b1   [4] W2 [32,32] [5] b2   [6] W3 [7] b3
//   [8] W4 [9] b4  [10] W5 [32,3] [11] b5
// Output: [N,3] float32.   Workspace: dinv(N) + hW(N*32) + h(N*32) = ~26 MB.
// ---------------------------------------------------------------------------
extern "C" void kernel_launch(void* const* d_in, const int* in_sizes, int n_in,
                              void* d_out, int out_size, void* d_ws, size_t ws_size,
                              hipStream_t stream) {
  const float* x  = (const float*)d_in[0];
  const int*   ei = (const int*)d_in[1];
  const int*   row = ei;             // edge_index[0]
  const int*   col = ei + N_EDGES;   // edge_index[1]

  const float* Wm[5] = {(const float*)d_in[2], (const float*)d_in[4],
                        (const float*)d_in[6], (const float*)d_in[8],
                        (const float*)d_in[10]};
  const float* Bm[5] = {(const float*)d_in[3], (const float*)d_in[5],
                        (const float*)d_in[7], (const float*)d_in[9],
                        (const float*)d_in[11]};

  float* ws   = (float*)d_ws;
  float* dinv = ws;                         // N floats
  float* hW   = ws + N_NODES;               // N*32 floats
  float* h    = ws + N_NODES + N_NODES * 32; // N*32 floats

  const int B = 256;
  const int gN  = (N_NODES + B - 1) / B;
  const int gE  = (N_EDGES + B - 1) / B;
  const int gGemm = ((N_NODES / 16) * 32 + B - 1) / B;  // 6250 waves

  // --- normalization ---
  deg_init_kernel<<<gN, B, 0, stream>>>(dinv);
  deg_count_kernel<<<gE, B, 0, stream>>>(dinv, col);
  deg_rsqrt_kernel<<<gN, B, 0, stream>>>(dinv);

  // --- layer 1: 8 -> 32 (input x) ---
  gemm_wmma_kernel<8, 32><<<gGemm, B, 0, stream>>>(x, Wm[0], hW);
  {
    long long n = (long long)N_NODES * 32;
    long long m = (long long)N_EDGES * 32;
    agg_init_kernel<32><<<(int)((n + B - 1) / B), B, 0, stream>>>(hW, Bm[0], dinv, h);
    agg_edge_kernel<32><<<(int)((m + B - 1) / B), B, 0, stream>>>(hW, row, col, dinv, h);
  }

  // --- layers 2..4: 32 -> 32 ---
  for (int l = 1; l <= 3; ++l) {
    gemm_wmma_kernel<32, 32><<<gGemm, B, 0, stream>>>(h, Wm[l], hW);
    long long n = (long long)N_NODES * 32;
    long long m = (long long)N_EDGES * 32;
    agg_init_kernel<32><<<(int)((n + B - 1) / B), B, 0, stream>>>(hW, Bm[l], dinv, h);
    agg_edge_kernel<32><<<(int)((m + B - 1) / B), B, 0, stream>>>(hW, row, col, dinv, h);
  }

  // --- layer 5: 32 -> 3, aggregate straight into d_out ---
  gemm_wmma_kernel<32, 3><<<gGemm, B, 0, stream>>>(h, Wm[4], hW);
  {
    float* out = (float*)d_out;
    long long n = (long long)N_NODES * 3;
    long long m = (long long)N_EDGES * 3;
    agg_init_kernel<3><<<(int)((n + B - 1) / B), B, 0, stream>>>(hW, Bm[4], dinv, out);
    agg_edge_kernel<3><<<(int)((m + B - 1) / B), B, 0, stream>>>(hW, row, col, dinv, out);
  }
}